// GraphLSTMCell_57604101374381
// MI455X (gfx1250) — compile-verified
//
#include <hip/hip_runtime.h>
#include <math.h>

#define BNODES 40000
#define NGRAPH 4
#define NNODES 10000
#define UNITS  32
#define GW     128
#define NEDGE  640000
#define TEDGE  (NEDGE + BNODES)

typedef __attribute__((ext_vector_type(2))) float v2f;
typedef __attribute__((ext_vector_type(8))) float v8f;

__device__ __forceinline__ float leaky_f(float x) { return x > 0.f ? x : 0.2f * x; }

__global__ void fill_kernel(float* __restrict__ p, float v, int n) {
  int i = blockIdx.x * blockDim.x + threadIdx.x;
  if (i < n) p[i] = v;
}

// Y[M,GW] = act(X[M,K] @ W[K,GW] + bias); one wave32 per 16x16 tile, fp32 WMMA.
// A frag (16x4 f32): lanes 0-15 -> M=l, K={k0,k0+1}; lanes 16-31 -> M=l, K={k0+2,k0+3}
// B frag (4x16 f32): lanes 0-15 -> N=l, K={k0,k0+1}; lanes 16-31 -> N=l, K={k0+2,k0+3}
__global__ void gemm_wmma_f32(const float* __restrict__ X,
                              const float* __restrict__ W,
                              const float* __restrict__ bias,
                              float* __restrict__ Y,
                              int K, int doLeaky) {
  const int tm   = blockIdx.x;
  const int tn   = blockIdx.y;
  const int lane = threadIdx.x;        // 0..31 (wave32)
  const int half = lane >> 4;
  const int l    = lane & 15;
  v8f c = {0.f, 0.f, 0.f, 0.f, 0.f, 0.f, 0.f, 0.f};
  const float* xrow = X + (size_t)(tm * 16 + l) * K;
  const float* wcol = W + tn * 16 + l;
  for (int k0 = 0; k0 < K; k0 += 4) {
    v2f a, b;
    a[0] = xrow[k0 + 2 * half];
    a[1] = xrow[k0 + 2 * half + 1];
    b[0] = wcol[(size_t)(k0 + 2 * half) * GW];
    b[1] = wcol[(size_t)(k0 + 2 * half + 1) * GW];
    c = __builtin_amdgcn_wmma_f32_16x16x4_f32(false, a, false, b, (short)0, c,
                                              false, false);
  }
  const float bv = bias ? bias[tn * 16 + l] : 0.f;
#pragma unroll
  for (int j = 0; j < 8; ++j) {
    int m = tm * 16 + j + 8 * half;
    float v = c[j] + bv;
    if (doLeaky) v = leaky_f(v);
    Y[(size_t)m * GW + tn * 16 + l] = v;
  }
}

__device__ __forceinline__ void atomicMaxF(float* addr, float val) {
  int* ai = (int*)addr;
  int old = *ai;
  while (__int_as_float(old) < val) {
    int assumed = old;
    old = atomicCAS(ai, assumed, __float_as_int(val));
    if (old == assumed) break;
  }
}

// Wave-per-edge: score = dot(Q[row], K[col]) over 128 dims; atomic segment max.
__global__ void edge_score(const int* __restrict__ ei,
                           const float* __restrict__ Q,
                           const float* __restrict__ Kmat,
                           float* __restrict__ score,
                           float* __restrict__ rowmax) {
  int wid  = (blockIdx.x * blockDim.x + threadIdx.x) >> 5;
  int lane = threadIdx.x & 31;
  if (wid >= TEDGE) return;
  int r, c;
  if (wid < NEDGE) { r = ei[wid]; c = ei[NEDGE + wid]; }
  else             { r = wid - NEDGE; c = r; }
  const float4 qa = ((const float4*)(Q    + (size_t)r * GW))[lane];
  const float4 ka = ((const float4*)(Kmat + (size_t)c * GW))[lane];
  float s = qa.x * ka.x + qa.y * ka.y + qa.z * ka.z + qa.w * ka.w;
#pragma unroll
  for (int off = 16; off > 0; off >>= 1) s += __shfl_xor(s, off, 32);
  if (lane == 0) { score[wid] = s; atomicMaxF(&rowmax[r], s); }
}

// Wave-per-edge: ex = exp(score - rowmax[row]); denom += ex; acc[row] += ex*V[col]
__global__ void edge_accum(const int* __restrict__ ei,
                           const float* __restrict__ V,
                           const float* __restrict__ score,
                           const float* __restrict__ rowmax,
                           float* __restrict__ denom,
                           float* __restrict__ acc) {
  int wid  = (blockIdx.x * blockDim.x + threadIdx.x) >> 5;
  int lane = threadIdx.x & 31;
  if (wid >= TEDGE) return;
  int r, c;
  if (wid < NEDGE) { r = ei[wid]; c = ei[NEDGE + wid]; }
  else             { r = wid - NEDGE; c = r; }
  float ex = expf(score[wid] - rowmax[r]);
  if (lane == 0) atomicAdd(&denom[r], ex);
  const float4 va = ((const float4*)(V + (size_t)c * GW))[lane];
  float* a = acc + (size_t)r * GW + lane * 4;
  atomicAdd(a + 0, ex * va.x);
  atomicAdd(a + 1, ex * va.y);
  atomicAdd(a + 2, ex * va.z);
  atomicAdd(a + 3, ex * va.w);
}

// sum += acc/denom + bv  (per-GAT finalize, accumulates the two GATs of a block)
__global__ void finalize_gat(const float* __restrict__ acc,
                             const float* __restrict__ denom,
                             const float* __restrict__ bv,
                             float* __restrict__ sum) {
  int idx = blockIdx.x * blockDim.x + threadIdx.x;
  if (idx >= BNODES * GW) return;
  int n = idx / GW, f = idx % GW;
  sum[idx] += acc[idx] / denom[n] + bv[f];
}

// Per (graph, feature) mean/std over NNODES nodes (population std, like jnp.std)
__global__ void gn_stats(const float* __restrict__ g, float* __restrict__ stats) {
  __shared__ float ss[256], ss2[256];
  const int f = blockIdx.x, b = blockIdx.y, tid = threadIdx.x;
  float s = 0.f, s2 = 0.f;
  for (int i = tid; i < NNODES; i += 256) {
    float v = g[(size_t)(b * NNODES + i) * GW + f];
    s += v; s2 += v * v;
  }
  ss[tid] = s; ss2[tid] = s2;
  __syncthreads();
  for (int o = 128; o > 0; o >>= 1) {
    if (tid < o) { ss[tid] += ss[tid + o]; ss2[tid] += ss2[tid + o]; }
    __syncthreads();
  }
  if (tid == 0) {
    float mean = ss[0] / (float)NNODES;
    float var  = ss2[0] / (float)NNODES - mean * mean;
    var = var > 0.f ? var : 0.f;
    stats[(b * GW + f) * 2 + 0] = mean;
    stats[(b * GW + f) * 2 + 1] = sqrtf(var);
  }
}

// in-place: g = leaky(gamma * (g-mean)/(std+eps) + beta)
__global__ void gn_apply(float* __restrict__ g,
                         const float* __restrict__ stats,
                         const float* __restrict__ gamma,
                         const float* __restrict__ beta) {
  int idx = blockIdx.x * blockDim.x + threadIdx.x;
  if (idx >= BNODES * GW) return;
  int n = idx / GW, f = idx % GW;
  int b = n / NNODES;
  float mean = stats[(b * GW + f) * 2 + 0];
  float sd   = stats[(b * GW + f) * 2 + 1];
  float xn = (g[idx] - mean) / (sd + 1e-5f);
  g[idx] = leaky_f(gamma[idx] * xn + beta[idx]);
}

// gate order from jnp.split: [f, o, i, g] chunks of 32
__global__ void lstm_kernel(const float* __restrict__ xg,
                            const float* __restrict__ hg,
                            const float* __restrict__ cell,
                            float* __restrict__ out) {
  int idx = blockIdx.x * blockDim.x + threadIdx.x;
  if (idx >= BNODES * UNITS) return;
  int n = idx / UNITS, j = idx % UNITS;
  const float* xr = xg + (size_t)n * GW;
  const float* hr = hg + (size_t)n * GW;
  float f  = 1.f / (1.f + expf(-(xr[j]      + hr[j])));
  float o  = 1.f / (1.f + expf(-(xr[32 + j] + hr[32 + j])));
  float ii = 1.f / (1.f + expf(-(xr[64 + j] + hr[64 + j])));
  float gg = tanhf(xr[96 + j] + hr[96 + j]);
  float cc = f * cell[idx] + ii * gg;
  float nh = o * tanhf(cc);
  out[idx] = cc;                      // c
  out[BNODES * UNITS + idx] = nh;     // new h
}

extern "C" void kernel_launch(void* const* d_in, const int* in_sizes, int n_in,
                              void* d_out, int out_size, void* d_ws, size_t ws_size,
                              hipStream_t stream) {
  const float* x    = (const float*)d_in[0];
  const float* h    = (const float*)d_in[1];
  const float* cell = (const float*)d_in[2];
  const int*   ei   = (const int*)d_in[3];

  float* ws     = (float*)d_ws;
  float* Q      = ws;
  float* Kb     = Q  + (size_t)BNODES * GW;
  float* V      = Kb + (size_t)BNODES * GW;
  float* acc    = V  + (size_t)BNODES * GW;
  float* xgates = acc + (size_t)BNODES * GW;
  float* hgates = xgates + (size_t)BNODES * GW;
  float* score  = hgates + (size_t)BNODES * GW;
  float* rowmax = score + TEDGE;
  float* denom  = rowmax + BNODES;
  float* stats  = denom + BNODES;

  const int fillNG = (BNODES * GW + 255) / 256;
  const int fillN  = (BNODES + 255) / 256;
  const int egrid  = (TEDGE + 7) / 8;   // 8 waves of 32 per 256-thread block

  auto run_gat = [&](const float* feat, int din, int pbase, float* sumbuf) {
    const float* wq = (const float*)d_in[pbase + 0];
    const float* bq = (const float*)d_in[pbase + 1];
    const float* wk = (const float*)d_in[pbase + 2];
    const float* bk = (const float*)d_in[pbase + 3];
    const float* wv = (const float*)d_in[pbase + 4];
    const float* bv = (const float*)d_in[pbase + 5];

    fill_kernel<<<fillN, 256, 0, stream>>>(rowmax, -3.0e38f, BNODES);
    fill_kernel<<<fillN, 256, 0, stream>>>(denom, 0.f, BNODES);
    fill_kernel<<<fillNG, 256, 0, stream>>>(acc, 0.f, BNODES * GW);

    dim3 gg(BNODES / 16, GW / 16);
    gemm_wmma_f32<<<gg, 32, 0, stream>>>(feat, wq, bq, Q, din, 1);
    gemm_wmma_f32<<<gg, 32, 0, stream>>>(feat, wk, bk, Kb, din, 1);
    gemm_wmma_f32<<<gg, 32, 0, stream>>>(feat, wv, nullptr, V, din, 0);

    edge_score<<<egrid, 256, 0, stream>>>(ei, Q, Kb, score, rowmax);
    edge_accum<<<egrid, 256, 0, stream>>>(ei, V, score, rowmax, denom, acc);
    finalize_gat<<<fillNG, 256, 0, stream>>>(acc, denom, bv, sumbuf);
  };

  auto run_block = [&](const float* feat, int din, int p1, int p2,
                       const float* gamma, const float* beta, float* gates) {
    fill_kernel<<<fillNG, 256, 0, stream>>>(gates, 0.f, BNODES * GW);
    run_gat(feat, din, p1, gates);
    run_gat(feat, din, p2, gates);
    gn_stats<<<dim3(GW, NGRAPH), 256, 0, stream>>>(gates, stats);
    gn_apply<<<fillNG, 256, 0, stream>>>(gates, stats, gamma, beta);
  };

  // x block: gx1 params at d_in[4..9], gx2 at d_in[10..15]; norms at 28/29
  run_block(x, 64, 4, 10, (const float*)d_in[28], (const float*)d_in[29], xgates);
  // h block: gh1 at d_in[16..21], gh2 at d_in[22..27]; norms at 30/31
  run_block(h, 32, 16, 22, (const float*)d_in[30], (const float*)d_in[31], hgates);

  lstm_kernel<<<(BNODES * UNITS + 255) / 256, 256, 0, stream>>>(
      xgates, hgates, cell, (float*)d_out);
}